// EmbeddingOHLinear_37117107372549
// MI455X (gfx1250) — compile-verified
//
#include <hip/hip_runtime.h>

// out[t, e] = W[e, X[t]] + b[e]   (embedding gather, NOT a GEMM)
// Pass 1: WTb[v][e] = W[e][v] + b[e]      (LDS-tiled transpose, bias fused)
// Pass 2: 16 tokens per workgroup via TDM:
//           TENSOR_LOAD_TO_LDS  (gather mode, 16 x 16-bit row indices) WTb -> LDS
//           TENSOR_STORE_FROM_LDS (normal mode, 16xE contiguous)       LDS -> out

typedef float v4f  __attribute__((ext_vector_type(4)));
typedef unsigned int u32x4 __attribute__((ext_vector_type(4)));
typedef int  i32x4 __attribute__((ext_vector_type(4)));
typedef int  i32x8 __attribute__((ext_vector_type(8)));

#define TT 32   // transpose tile
#define TY 8    // 32x8 = 256 threads = 8 wave32
#define TPB 16  // tokens per gather workgroup (max rows per 16-bit-index TDM gather)
#define EMAX 1024

// ---------------------------------------------------------------- transpose
__global__ void ohlin_transpose_bias(const float* __restrict__ W,
                                     const float* __restrict__ b,
                                     float* __restrict__ WTb,
                                     int E, int V) {
  __shared__ float tile[TT][TT + 1];
  const int v0 = blockIdx.x * TT;
  const int e0 = blockIdx.y * TT;
  const int x  = threadIdx.x;
  const int y  = threadIdx.y;
  #pragma unroll
  for (int j = 0; j < TT; j += TY)
    tile[y + j][x] = W[(size_t)(e0 + y + j) * (size_t)V + (size_t)(v0 + x)];
  __syncthreads();
  const float bias = b[e0 + x];
  #pragma unroll
  for (int j = 0; j < TT; j += TY)
    WTb[(size_t)(v0 + y + j) * (size_t)E + (size_t)(e0 + x)] = tile[x][y + j] + bias;
}

// ---------------------------------------------------------------- TDM gather
#ifndef __has_builtin
#define __has_builtin(x) 0
#endif

#if defined(__AMDGCN__) && __has_builtin(__builtin_amdgcn_tensor_load_to_lds) && \
    __has_builtin(__builtin_amdgcn_tensor_store_from_lds) && \
    __has_builtin(__builtin_amdgcn_s_wait_tensorcnt)
#define OHLIN_HAVE_TDM 1
#else
#define OHLIN_HAVE_TDM 0
#endif

#if OHLIN_HAVE_TDM
#if defined(__clang_major__) && (__clang_major__ >= 23)
#define TDM_LOAD(g0, g1, g2, g3)  __builtin_amdgcn_tensor_load_to_lds((g0), (g1), (g2), (g3), (i32x8)0, 0)
#define TDM_STORE(g0, g1, g2, g3) __builtin_amdgcn_tensor_store_from_lds((g0), (g1), (g2), (g3), (i32x8)0, 0)
#else
#define TDM_LOAD(g0, g1, g2, g3)  __builtin_amdgcn_tensor_load_to_lds((g0), (g1), (g2), (g3), 0)
#define TDM_STORE(g0, g1, g2, g3) __builtin_amdgcn_tensor_store_from_lds((g0), (g1), (g2), (g3), 0)
#endif

// D# group 0: count=1 (valid user descriptor), optional gather flags,
// LDS byte address in [63:32], global byte address in [120:64], type=2 in [127:126].
__device__ __forceinline__ u32x4 tdm_group0(unsigned flags, unsigned lds_addr,
                                            unsigned long long ga) {
  u32x4 g0;
  g0.x = 1u | flags;                                          // count=1 | gather bits
  g0.y = lds_addr;                                            // lds_addr
  g0.z = (unsigned)(ga & 0xffffffffu);                        // global_addr[31:0]
  g0.w = (unsigned)((ga >> 32) & 0x01ffffffu) | 0x80000000u;  // addr[56:32] | type=2
  return g0;
}

// D# group 1 for a (tile_dim0=E) x (tile_dim1=TPB) tile of 4-byte elements,
// row pitch tensor_dim0_stride=E, tensor dims E x dim1 (OOB bounds).
__device__ __forceinline__ i32x8 tdm_group1(unsigned E, unsigned dim1) {
  i32x8 g1;
  g1[0] = (int)(2u << 16);                                      // data_size=4B, mask/pad/iter=0
  g1[1] = (int)((E & 0xffffu) << 16);                           // tensor_dim0[15:0]
  g1[2] = (int)((E >> 16) | ((dim1 & 0xffffu) << 16));          // tensor_dim0[31:16] | tensor_dim1[15:0]
  g1[3] = (int)(((dim1 >> 16) & 0xffffu) | ((E & 0xffffu) << 16)); // tensor_dim1[31:16] | tile_dim0
  g1[4] = (int)(TPB & 0xffffu);                                 // tile_dim1 (= #gather indices)
  g1[5] = (int)E;                                               // tensor_dim0_stride[31:0]
  g1[6] = 0;                                                    // stride0[47:32] | dim1_stride[15:0]
  g1[7] = 0;                                                    // tensor_dim1_stride[47:16]
  return g1;
}
#endif  // OHLIN_HAVE_TDM

// One wave per TPB tokens. Gather-load 16 WTb rows into LDS with one TDM
// descriptor, then store the contiguous 16xE block to out with a second one.
__global__ void __launch_bounds__(32)
ohlin_gather_tdm(const int* __restrict__ X,
                 const float* __restrict__ WTb,
                 float* __restrict__ out,
                 int E, int V) {
#if OHLIN_HAVE_TDM
  __shared__ float rows[TPB * EMAX];               // 64 KB staging
  const int base = blockIdx.x * TPB;

  // Pack 16 row indices (uniform scalar loads) into D# groups 2/3, 16-bit mode.
  unsigned id[TPB];
  #pragma unroll
  for (int i = 0; i < TPB; ++i)
    id[i] = ((unsigned)X[base + i]) & 0xffffu;
  i32x4 g2 = { (int)(id[0]  | (id[1]  << 16)), (int)(id[2]  | (id[3]  << 16)),
               (int)(id[4]  | (id[5]  << 16)), (int)(id[6]  | (id[7]  << 16)) };
  i32x4 g3 = { (int)(id[8]  | (id[9]  << 16)), (int)(id[10] | (id[11] << 16)),
               (int)(id[12] | (id[13] << 16)), (int)(id[14] | (id[15] << 16)) };

  const unsigned lds = (unsigned)(size_t)&rows[0];
  const unsigned long long gsrc = (unsigned long long)(size_t)WTb;   // table base; rows via indices
  const unsigned long long gdst =
      (unsigned long long)(size_t)(out + (size_t)base * (size_t)E);

  // gather_mode=1 (bit31), 16-bit indices (bit30=0); tensor_dim1=V bounds the row index.
  const u32x4 g0_ld = tdm_group0(0x80000000u, lds, gsrc);
  const i32x8 g1_ld = tdm_group1((unsigned)E, (unsigned)V);
  TDM_LOAD(g0_ld, g1_ld, g2, g3);

  __builtin_amdgcn_s_wait_tensorcnt(0);            // rows resident in LDS

  // Normal-mode contiguous store: 16 x E tile, tensor dims E x 16, pitch E.
  const u32x4 g0_st = tdm_group0(0u, lds, gdst);
  const i32x8 g1_st = tdm_group1((unsigned)E, (unsigned)TPB);
  i32x4 z4 = (i32x4)0;
  TDM_STORE(g0_st, g1_st, z4, z4);
  // S_ENDPGM performs an implicit wait-idle, covering the outstanding store.
#else
  // Portable fallback (host pass / toolchains without the TDM builtins):
  // plain coalesced float4 row copies, one wave doing TPB rows.
  const int base = blockIdx.x * TPB;
  const int Ev4  = E >> 2;
  const v4f* __restrict__ src4 = (const v4f*)WTb;
  v4f* __restrict__ dst4 = (v4f*)out;
  for (int k = 0; k < TPB; ++k) {
    const int r = X[base + k];
    for (int t = threadIdx.x; t < Ev4; t += 32) {
      v4f v = src4[(size_t)r * (size_t)Ev4 + (size_t)t];
      __builtin_nontemporal_store(v, &dst4[(size_t)(base + k) * (size_t)Ev4 + (size_t)t]);
    }
  }
  (void)V;
#endif
}

// ---------------------------------------------------------------- fallback
__global__ void ohlin_direct(const int* __restrict__ X,
                             const float* __restrict__ W,
                             const float* __restrict__ b,
                             float* __restrict__ out,
                             int E, int V) {
  const int token = blockIdx.x;
  const int idx   = X[token];
  for (int e = threadIdx.x; e < E; e += blockDim.x) {
    float v = W[(size_t)e * (size_t)V + (size_t)idx] + b[e];
    __builtin_nontemporal_store(v, &out[(size_t)token * (size_t)E + (size_t)e]);
  }
}

// ---------------------------------------------------------------- launcher
extern "C" void kernel_launch(void* const* d_in, const int* in_sizes, int n_in,
                              void* d_out, int out_size, void* d_ws, size_t ws_size,
                              hipStream_t stream) {
  const int*   X = (const int*)d_in[0];     // [16, 2048] int32 labels
  const float* W = (const float*)d_in[1];   // [E, V] row-major
  const float* b = (const float*)d_in[2];   // [E]
  float*     out = (float*)d_out;           // [T, E]

  const int T = in_sizes[0];                // 32768 tokens
  const int E = in_sizes[2];                // 1024
  const int V = in_sizes[1] / E;            // 8192

  const size_t need = (size_t)E * (size_t)V * sizeof(float);  // 32 MB
  const bool fast = (ws_size >= need) &&
                    (E % TT == 0) && (V % TT == 0) && (E % 4 == 0) &&
                    (E <= EMAX) &&            // 64 KB LDS staging bound
                    (V <= 65536) &&           // 16-bit gather indices
                    (T % TPB == 0);

  if (fast) {
    float* WTb = (float*)d_ws;
    dim3 tb(TT, TY, 1);
    dim3 tg(V / TT, E / TT, 1);
    ohlin_transpose_bias<<<tg, tb, 0, stream>>>(W, b, WTb, E, V);
    // Same stream => transpose completes before the gather reads WTb.
    ohlin_gather_tdm<<<dim3(T / TPB, 1, 1), dim3(32, 1, 1), 0, stream>>>(
        X, WTb, out, E, V);
  } else {
    ohlin_direct<<<dim3(T, 1, 1), dim3(256, 1, 1), 0, stream>>>(X, W, b, out, E, V);
  }
}